// BayesianModel_21449066676258
// MI455X (gfx1250) — compile-verified
//
#include <hip/hip_runtime.h>
#include <hip/hip_bf16.h>
#include <math.h>

typedef __attribute__((ext_vector_type(2))) float v2f;
typedef __attribute__((ext_vector_type(8))) float v8f;

namespace {
constexpr int T_DIM = 1024;
constexpr int B_DIM = 2048;
constexpr int F_DIM = 11;
constexpr int G_DIM = 100;
constexpr float KAPPA = 0.1f;
constexpr int SEQ_PER_WAVE = 16;   // WMMA N dimension = sequences
constexpr int WAVES_PER_BLOCK = 2;
constexpr int THREADS = WAVES_PER_BLOCK * 32;
constexpr int NCHUNK = G_DIM / 4;  // 25 K-chunks of 4 for V_WMMA_F32_16X16X4_F32
constexpr int LDS_SEQ_PAD = 9;     // 9-float row pad -> conflict-free ds reads
}

#if defined(__HIP_DEVICE_COMPILE__) &&                                        \
    __has_builtin(__builtin_amdgcn_global_load_async_to_lds_b32) &&           \
    __has_builtin(__builtin_amdgcn_s_wait_asynccnt)
#define USE_ASYNC_LDS 1
typedef __attribute__((address_space(1))) int gas_int;  // global (AS1)
typedef __attribute__((address_space(3))) int las_int;  // LDS (AS3)
#else
#define USE_ASYNC_LDS 0
#endif

#if defined(__HIP_DEVICE_COMPILE__) &&                                        \
    __has_builtin(__builtin_amdgcn_wmma_f32_16x16x4_f32)
#define USE_WMMA 1
#else
#define USE_WMMA 0
#endif

__device__ __forceinline__ float clamp01(float x) {
  return fminf(fmaxf(x, 0.0f), 1.0f);
}
__device__ __forceinline__ float nan0(float x) { return (x == x) ? x : 0.0f; }

__global__ __launch_bounds__(THREADS) void bayes_scan_wmma_kernel(
    const float* __restrict__ inputs,     // (T,1,B,11)
    const float* __restrict__ beta_raw,   // (P,)
    const float* __restrict__ sigma_raw,  // (P,)
    float* __restrict__ out_logits,       // (T,1,B,2)
    float* __restrict__ out_belief) {     // (B,100)
  const int lane = threadIdx.x & 31;
  const int wave = threadIdx.x >> 5;
  const int m = lane & 15;     // sequence slot (WMMA column / A row)
  const int half = lane >> 4;  // K-offset group: 0 -> {4c,4c+1}, 1 -> {4c+2,4c+3}
  const int b0 = (blockIdx.x * WAVES_PER_BLOCK + wave) * SEQ_PER_WAVE;
  const int b = b0 + m;  // this lane's sequence (duplicated in both halves)

#if USE_ASYNC_LDS
  __shared__ float lbuf[WAVES_PER_BLOCK][2][SEQ_PER_WAVE * LDS_SEQ_PAD];
#endif

  // Belief storage: this lane holds B-matrix entries bel[k][m] for
  // k = 4c + 2*half + {0,1}, c = 0..24  (exactly 100 = 25*4, no masking).
  float bel[2 * NCHUNK];
#pragma unroll
  for (int i = 0; i < 2 * NCHUNK; ++i) bel[i] = 1.0f;

  // Running p_mu for this lane's two k slots, stepped by 4/99 per chunk.
  const float pstep = 4.0f / 99.0f;
  const float pm0_0 = (float)(2 * half) * (1.0f / 99.0f);
  const float pm1_0 = (float)(2 * half + 1) * (1.0f / 99.0f);
  // A-matrix (16x4 constant): row 0 = ones (S), row 1 = p_mu chunk (E-moment).
  float aA0 = 0.0f, aB0 = 0.0f, ainc = 0.0f;
  if (m == 0) { aA0 = 1.0f; aB0 = 1.0f; }
  if (m == 1) { aA0 = pm0_0; aB0 = pm1_0; ainc = pstep; }

  // Per-sequence parameters (constant over the scan).
  const int pid = (int)nan0(inputs[(size_t)b * F_DIM + 10]);
  const float br = beta_raw[pid];
  const float sr = sigma_raw[pid];
  float beta = log1pf(expf(br)) + 1.0f;  // softplus + 1
  beta = fminf(fmaxf(beta, 1.0f), 25.0f);
  float sig = 1.0f / (1.0f + expf(-sr));  // sigmoid
  sig = fminf(fmaxf(sig * 0.09f + 0.01f, 0.01f), 0.1f);
  const float inv_s = 1.0f / (sig * 1.4142135623730951f);

  const size_t tstride = (size_t)B_DIM * F_DIM;

#if USE_ASYNC_LDS
  // 16 seq x 6 fields = 96 words = 3 async b32 issues with all 32 lanes live.
  const float* srcp[3];
  int didx[3];
#pragma unroll
  for (int e2 = 0; e2 < 3; ++e2) {
    const int idx = lane + 32 * e2;  // 0..95
    const int sq = idx / 6;
    const int fld = idx - 6 * sq;
    srcp[e2] = inputs + (size_t)(b0 + sq) * F_DIM + fld;
    didx[e2] = sq * LDS_SEQ_PAD + fld;
  }
  // Prologue: async-fetch step 0 into buffer 0.
#pragma unroll
  for (int e2 = 0; e2 < 3; ++e2)
    __builtin_amdgcn_global_load_async_to_lds_b32(
        (gas_int*)srcp[e2], (las_int*)&lbuf[wave][0][didx[e2]], 0, 0);
#endif

  int cur = 0;
#pragma unroll 1
  for (int t = 0; t < T_DIM; ++t) {
    // Warm L2 ahead of the scan (global_prefetch_b8).
    if (lane == 0 && t + 16 < T_DIM)
      __builtin_prefetch(inputs + (size_t)(t + 16) * tstride +
                             (size_t)b0 * F_DIM, 0, 0);

#if USE_ASYNC_LDS
    if (t + 1 < T_DIM) {
      const size_t off = (size_t)(t + 1) * tstride;
#pragma unroll
      for (int e2 = 0; e2 < 3; ++e2)
        __builtin_amdgcn_global_load_async_to_lds_b32(
            (gas_int*)(srcp[e2] + off),
            (las_int*)&lbuf[wave][cur ^ 1][didx[e2]], 0, 0);
      // In-order completion: <=3 outstanding => step-t's 3 loads retired.
      __builtin_amdgcn_s_wait_asynccnt(3);
    } else {
      __builtin_amdgcn_s_wait_asynccnt(0);
    }
    asm volatile("" ::: "memory");
    const float* sb = &lbuf[wave][cur][m * LDS_SEQ_PAD];
    const float act0 = nan0(sb[0]);
    const float act1 = nan0(sb[1]);
    const float rew0 = nan0(sb[2]);
    const float rew1 = nan0(sb[3]);
    const float cdc  = nan0(sb[4]);
    const float cdn  = nan0(sb[5]);
#else
    const float* src = inputs + (size_t)t * tstride + (size_t)b * F_DIM;
    const float act0 = nan0(src[0]);
    const float act1 = nan0(src[1]);
    const float rew0 = nan0(src[2]);
    const float rew1 = nan0(src[3]);
    const float cdc  = nan0(src[4]);
    const float cdn  = nan0(src[5]);
#endif

    const bool had = (act0 + act1) > 0.0f;
    const float r_s = fmaf(rew0, act0, rew1 * act1);
    const float r = (act1 > act0) ? (1.0f - r_s) : r_s;  // argmax ties -> 0

    // state_beliefs(cd_current)
    const float u = 0.5f * (1.0f + erff((0.0f - cdc) * inv_s));
    const float v = 0.5f * (1.0f + erff((-KAPPA - cdc) * inv_s));
    const float w = 0.5f * (1.0f + erff((KAPPA - cdc) * inv_s));
    const float den = fmaxf(w - v, 1e-10f);
    const float pi0 = clamp01((u - v) / den);
    const float pi1 = clamp01((w - u) / den);
    float q0 = fmaf(pi1, r, pi0 * (1.0f - r));
    float q1 = (2.0f * r - 1.0f) * (pi0 - pi1);
    if (!had) { q0 = 1.0f; q1 = 0.0f; }  // update = 1 everywhere

    // Belief update + [S; E] = [ones; p_mu] (2x100) @ bel (100x16) via
    // 25 chained V_WMMA_F32_16X16X4_F32 over 4 accumulators.
#if USE_WMMA
    const v8f vz = {0.f, 0.f, 0.f, 0.f, 0.f, 0.f, 0.f, 0.f};
    v8f acc[4] = {vz, vz, vz, vz};
#else
    float s_l = 0.0f, e_l = 0.0f;
#endif
    float cpm0 = pm0_0, cpm1 = pm1_0, caA = aA0, caB = aB0;
#pragma unroll
    for (int c = 0; c < NCHUNK; ++c) {
      const float up0 = fmaf(q1, cpm0, q0);
      const float up1 = fmaf(q1, cpm1, q0);
      const float nb0 = fmaxf(bel[2 * c] * up0, 1e-30f);
      const float nb1 = fmaxf(bel[2 * c + 1] * up1, 1e-30f);
      bel[2 * c] = nb0;
      bel[2 * c + 1] = nb1;
#if USE_WMMA
      v2f aV; aV.x = caA; aV.y = caB;   // A: 16x4 [ones; pmu; 0...]
      v2f bV; bV.x = nb0; bV.y = nb1;   // B: 4x16 belief chunk, N = sequence
      acc[c & 3] = __builtin_amdgcn_wmma_f32_16x16x4_f32(
          false, aV, false, bV, (short)0, acc[c & 3], false, false);
#else
      s_l += nb0 + nb1;
      e_l = fmaf(nb0, cpm0, fmaf(nb1, cpm1, e_l));
#endif
      cpm0 += pstep; cpm1 += pstep;
      caA += ainc;  caB += ainc;
    }
#if USE_WMMA
    // D row M=0 (VGPR0) = S[m] at lane m; row M=1 (VGPR1) = E-moment[m].
    const float S  = (acc[0][0] + acc[1][0]) + (acc[2][0] + acc[3][0]);
    const float Ew = (acc[0][1] + acc[1][1]) + (acc[2][1] + acc[3][1]);
    const float ratio = Ew / S;            // valid in lanes 0..15
    const float E = __shfl(ratio, m, 32);  // broadcast to both halves
#else
    const float S = s_l + __shfl_xor(s_l, 16, 32);   // halves partition k
    const float Es = e_l + __shfl_xor(e_l, 16, 32);
    const float E = Es / S;
#endif

    // state_beliefs(cd_next) -> action values -> logits
    const float un = 0.5f * (1.0f + erff((0.0f - cdn) * inv_s));
    const float vn = 0.5f * (1.0f + erff((-KAPPA - cdn) * inv_s));
    const float wn = 0.5f * (1.0f + erff((KAPPA - cdn) * inv_s));
    const float denn = fmaxf(wn - vn, 1e-10f);
    const float pi0n = clamp01((un - vn) / denn);
    const float pi1n = clamp01((wn - un) / denn);
    const float va0 = fmaf(pi0n - pi1n, E, pi1n);
    const float va1 = fmaf(pi1n - pi0n, E, pi0n);

    if (lane < 16) {  // one b64 store covering all 16 sequences
      float2 lg;
      lg.x = beta * va0;
      lg.y = beta * va1;
      *reinterpret_cast<float2*>(out_logits + ((size_t)t * B_DIM + b) * 2) = lg;
    }
    cur ^= 1;
  }

  // Final (unnormalized, clamped) belief carry: lane writes its k pairs.
  float* ob = out_belief + (size_t)b * G_DIM + 2 * half;
#pragma unroll
  for (int c = 0; c < NCHUNK; ++c) {
    float2 st;
    st.x = bel[2 * c];
    st.y = bel[2 * c + 1];
    *reinterpret_cast<float2*>(ob + 4 * c) = st;
  }
}

extern "C" void kernel_launch(void* const* d_in, const int* in_sizes, int n_in,
                              void* d_out, int out_size, void* d_ws,
                              size_t ws_size, hipStream_t stream) {
  (void)in_sizes; (void)n_in; (void)d_ws; (void)ws_size; (void)out_size;
  const float* inputs    = (const float*)d_in[0];
  const float* beta_raw  = (const float*)d_in[1];
  const float* sigma_raw = (const float*)d_in[2];
  float* out        = (float*)d_out;
  float* out_logits = out;                              // T*1*B*2
  float* out_belief = out + (size_t)T_DIM * B_DIM * 2;  // B*100

  // 2048 sequences / 16 per wave / 2 waves per block = 64 blocks.
  dim3 grid(B_DIM / (SEQ_PER_WAVE * WAVES_PER_BLOCK));
  dim3 block(THREADS);
  bayes_scan_wmma_kernel<<<grid, block, 0, stream>>>(inputs, beta_raw,
                                                     sigma_raw, out_logits,
                                                     out_belief);
}